// DBScan_79654463472059
// MI455X (gfx1250) — compile-verified
//
#include <hip/hip_runtime.h>
#include <stdint.h>

#define NPTS 8192
#define WORDS 256            // NPTS / 32
#define NUM_CLASSES 5
#define EPS2 225.0f          // 15^2
#define MIN_POINTS 5
#define SENT NPTS
#define BIGC (1 << 30)
#define PROP_ITERS 16

typedef float v2f __attribute__((ext_vector_type(2)));
typedef float v8f __attribute__((ext_vector_type(8)));

// ---------------------------------------------------------------- preprocess
__global__ void k_prep(const float* __restrict__ x, float4* __restrict__ P4,
                       float* __restrict__ p2, int* __restrict__ group) {
  int i = blockIdx.x * blockDim.x + threadIdx.x;
  if (i >= NPTS) return;
  const float* row = x + i * 10;
  float px = row[0], py = row[1], pz = row[2];
  int b = (int)row[3];
  float best = row[5];
  int cls = 0;
#pragma unroll
  for (int k = 1; k < NUM_CLASSES; ++k) {
    float v = row[5 + k];
    if (v > best) { best = v; cls = k; }      // first max, like jnp.argmax
  }
  P4[i] = make_float4(px, py, pz, 0.0f);
  p2[i] = px * px + py * py + pz * pz;
  group[i] = b * NUM_CLASSES + cls;
}

// ------------------------------------------------- adjacency bitmask via WMMA
// One wave computes a 16(i) x 32(j) patch of the Gram matrix with two
// V_WMMA_F32_16X16X4_F32 ops, then ballots the (same-group && d2<eps^2)
// predicate into one 32-bit adjacency word per row.
__global__ void k_adj(const float4* __restrict__ P4, const float* __restrict__ p2,
                      const int* __restrict__ group, unsigned* __restrict__ adj) {
  int lane = threadIdx.x & 31;
  int wave = (blockIdx.x * blockDim.x + threadIdx.x) >> 5;
  int iTile = wave >> 8;           // 512 i-tiles of 16 rows
  int jPair = wave & 255;          // 256 j-words of 32 cols
  int i0 = iTile * 16;
  int j0 = jPair * 32;

  int half = lane >> 4;            // 0 -> (K0,K1) ; 1 -> (K2,K3)
  int lm = lane & 15;

  // A operand: row i0+lm, K slice per half. K=3 padded with 0.
  float4 pa = P4[i0 + lm];
  v2f a = { half ? pa.z : pa.x, half ? 0.0f : pa.y };
  // B operands: columns j0+lm and j0+16+lm (column-major mirror of A layout)
  float4 pb0 = P4[j0 + lm];
  float4 pb1 = P4[j0 + 16 + lm];
  v2f b0 = { half ? pb0.z : pb0.x, half ? 0.0f : pb0.y };
  v2f b1 = { half ? pb1.z : pb1.x, half ? 0.0f : pb1.y };

  v8f cA = {};
  v8f cB = {};
  cA = __builtin_amdgcn_wmma_f32_16x16x4_f32(false, a, false, b0, (short)0, cA, false, false);
  cB = __builtin_amdgcn_wmma_f32_16x16x4_f32(false, a, false, b1, (short)0, cB, false, false);

  int jA = j0 + lm;
  int jB = j0 + 16 + lm;
  float p2jA = p2[jA], p2jB = p2[jB];
  int gjA = group[jA], gjB = group[jB];

#pragma unroll
  for (int r = 0; r < 8; ++r) {
    // C/D layout: VGPR r holds row m=r (lanes 0-15) and m=r+8 (lanes 16-31)
    int im = i0 + r + (half << 3);
    float p2i = p2[im];
    int gi = group[im];
    float d2A = p2i + p2jA - 2.0f * cA[r];
    float d2B = p2i + p2jB - 2.0f * cB[r];
    unsigned mA = __builtin_amdgcn_ballot_w32((gi == gjA) && (d2A < EPS2));
    unsigned mB = __builtin_amdgcn_ballot_w32((gi == gjB) && (d2B < EPS2));
    if (lane == 0) {
      adj[(size_t)(i0 + r) * WORDS + jPair]     = (mA & 0xFFFFu) | (mB << 16);
      adj[(size_t)(i0 + r + 8) * WORDS + jPair] = (mA >> 16) | (mB & 0xFFFF0000u);
    }
  }
}

// ---------------------------------------------- core flags + label init
__global__ void k_core(const unsigned* __restrict__ adj,
                       unsigned* __restrict__ coremask, int* __restrict__ lab) {
  int i = blockIdx.x * blockDim.x + threadIdx.x;
  const unsigned* row = adj + (size_t)i * WORDS;
  int cnt = 0;
  for (int w = 0; w < WORDS; ++w) cnt += __popc(row[w]);
  bool core = cnt >= MIN_POINTS;                 // diagonal included, as in ref
  unsigned m = __builtin_amdgcn_ballot_w32(core);
  if ((threadIdx.x & 31) == 0) coremask[i >> 5] = m;
  lab[i] = core ? i : SENT;
}

// ---------------------------------------------- one wave per point: min over
// core-core neighbors of lab[j]
__global__ void k_neighmin(const unsigned* __restrict__ adj,
                           const unsigned* __restrict__ coremask,
                           const int* __restrict__ lab, int* __restrict__ tmp) {
  int lane = threadIdx.x & 31;
  int i = (blockIdx.x * blockDim.x + threadIdx.x) >> 5;
  bool core = (coremask[i >> 5] >> (i & 31)) & 1;   // wave-uniform
  int mn = SENT;
  if (core) {
    const unsigned* row = adj + (size_t)i * WORDS;
    for (int w = lane; w < WORDS; w += 32) {
      unsigned m = row[w] & coremask[w];
      int base = w * 32;
      while (m) {
        int b = __builtin_ctz(m);
        m &= m - 1;
        int lj = lab[base + b];
        mn = mn < lj ? mn : lj;
      }
    }
  }
  for (int off = 16; off; off >>= 1) {
    int o = __shfl_xor(mn, off, 32);
    mn = mn < o ? mn : o;
  }
  if (lane == 0) {
    int li = lab[i];
    tmp[i] = core ? (mn < li ? mn : li) : SENT;
  }
}

// ---------------------------------------------- pointer-jump shortcut
__global__ void k_shortcut(const unsigned* __restrict__ coremask,
                           const int* __restrict__ tmp, int* __restrict__ lab) {
  int i = blockIdx.x * blockDim.x + threadIdx.x;
  bool core = (coremask[i >> 5] >> (i & 31)) & 1;
  if (core) {
    int a = tmp[i];
    int t = tmp[a];
    lab[i] = t < a ? t : a;
  } else {
    lab[i] = SENT;
  }
}

__global__ void k_isrep(const unsigned* __restrict__ coremask,
                        const int* __restrict__ lab, int* __restrict__ isrep) {
  int i = blockIdx.x * blockDim.x + threadIdx.x;
  bool core = (coremask[i >> 5] >> (i & 31)) & 1;
  isrep[i] = (core && lab[i] == i) ? 1 : 0;
}

// cluster id = #reps in same group with smaller index (matches cumsum order)
__global__ void k_cid(const int* __restrict__ isrep, const int* __restrict__ group,
                      int* __restrict__ cidrep) {
  int i = blockIdx.x * blockDim.x + threadIdx.x;
  int c = 0;
  if (isrep[i]) {
    int g = group[i];
    for (int j = 0; j < i; ++j) c += (isrep[j] && group[j] == g) ? 1 : 0;
  }
  cidrep[i] = c;
}

__global__ void k_ccid(const unsigned* __restrict__ coremask,
                       const int* __restrict__ lab, const int* __restrict__ cidrep,
                       int* __restrict__ ccid) {
  int i = blockIdx.x * blockDim.x + threadIdx.x;
  bool core = (coremask[i >> 5] >> (i & 31)) & 1;
  ccid[i] = core ? cidrep[lab[i]] : BIGC;
}

// ---------------------------------------------- final labels + masked copy
__global__ void k_final(const unsigned* __restrict__ adj,
                        const unsigned* __restrict__ coremask,
                        const int* __restrict__ ccid, const float* __restrict__ x,
                        float* __restrict__ out) {
  int lane = threadIdx.x & 31;
  int i = (blockIdx.x * blockDim.x + threadIdx.x) >> 5;
  bool core = (coremask[i >> 5] >> (i & 31)) & 1;   // wave-uniform
  int label;
  if (core) {
    label = ccid[i];
  } else {
    int mn = BIGC;
    const unsigned* row = adj + (size_t)i * WORDS;
    for (int w = lane; w < WORDS; w += 32) {
      unsigned m = row[w] & coremask[w];
      int base = w * 32;
      while (m) {
        int b = __builtin_ctz(m);
        m &= m - 1;
        int c = ccid[base + b];
        mn = mn < c ? mn : c;
      }
    }
    for (int off = 16; off; off >>= 1) {
      int o = __shfl_xor(mn, off, 32);
      mn = mn < o ? mn : o;
    }
    label = (mn == BIGC) ? -1 : mn;
  }
  if (lane == 0) {
    out[i] = (float)label;
    float* crow = out + NPTS + i * 5;
    const float* xr = x + i * 10;
#pragma unroll
    for (int k = 0; k < 5; ++k) crow[k] = (label >= 0) ? xr[k] : 0.0f;
  }
}

extern "C" void kernel_launch(void* const* d_in, const int* in_sizes, int n_in,
                              void* d_out, int out_size, void* d_ws, size_t ws_size,
                              hipStream_t stream) {
  (void)in_sizes; (void)n_in; (void)out_size; (void)ws_size;
  const float* x = (const float*)d_in[0];
  float* out = (float*)d_out;

  char* ws = (char*)d_ws;
  size_t off = 0;
  auto alloc = [&](size_t bytes) -> void* {
    void* p = ws + off;
    off = (off + bytes + 255) & ~(size_t)255;
    return p;
  };
  float4*   P4       = (float4*)alloc((size_t)NPTS * sizeof(float4));
  float*    p2       = (float*)alloc((size_t)NPTS * 4);
  int*      group    = (int*)alloc((size_t)NPTS * 4);
  unsigned* adj      = (unsigned*)alloc((size_t)NPTS * WORDS * 4);  // 8 MB
  unsigned* coremask = (unsigned*)alloc((size_t)WORDS * 4);
  int*      lab      = (int*)alloc((size_t)NPTS * 4);
  int*      tmp      = (int*)alloc((size_t)NPTS * 4);
  int*      isrep    = (int*)alloc((size_t)NPTS * 4);
  int*      cidrep   = (int*)alloc((size_t)NPTS * 4);
  int*      ccid     = (int*)alloc((size_t)NPTS * 4);

  k_prep<<<NPTS / 256, 256, 0, stream>>>(x, P4, p2, group);

  // 512 i-tiles * 256 j-words = 131072 waves, 8 waves per 256-thread block
  int adjBlocks = (NPTS / 16) * (NPTS / 32) / 8;
  k_adj<<<adjBlocks, 256, 0, stream>>>(P4, p2, group, adj);

  k_core<<<NPTS / 256, 256, 0, stream>>>(adj, coremask, lab);

  for (int it = 0; it < PROP_ITERS; ++it) {
    k_neighmin<<<NPTS * 32 / 256, 256, 0, stream>>>(adj, coremask, lab, tmp);
    k_shortcut<<<NPTS / 256, 256, 0, stream>>>(coremask, tmp, lab);
  }

  k_isrep<<<NPTS / 256, 256, 0, stream>>>(coremask, lab, isrep);
  k_cid<<<NPTS / 256, 256, 0, stream>>>(isrep, group, cidrep);
  k_ccid<<<NPTS / 256, 256, 0, stream>>>(coremask, lab, cidrep, ccid);
  k_final<<<NPTS * 32 / 256, 256, 0, stream>>>(adj, coremask, ccid, x, out);
}